// LabelGNN_30709016166501
// MI455X (gfx1250) — compile-verified
//
#include <hip/hip_runtime.h>
#include <hip/hip_bf16.h>
#include <math.h>

#define GD 128              // feature dimension D
#define GAT_NEG_SLOPE 0.2f

typedef __attribute__((ext_vector_type(2))) float v2f;
typedef __attribute__((ext_vector_type(8))) float v8f;

// ---- monotone float <-> ordered-u32 mapping for atomic max on floats ----
__device__ __forceinline__ unsigned gat2_f2ord(float f) {
  unsigned u = __float_as_uint(f);
  return (u & 0x80000000u) ? ~u : (u | 0x80000000u);
}
__device__ __forceinline__ float gat2_ord2f(unsigned u) {
  return __uint_as_float((u & 0x80000000u) ? (u & 0x7FFFFFFFu) : ~u);
}

// ------------------------- init kernels -------------------------
__global__ void gat2_fill_u32(unsigned* __restrict__ p, unsigned v, int n) {
  int t = blockIdx.x * blockDim.x + threadIdx.x;
  if (t < n) p[t] = v;
}
__global__ void gat2_zero_f32(float* __restrict__ p, int n) {
  int t = blockIdx.x * blockDim.x + threadIdx.x;
  if (t < n) p[t] = 0.0f;
}

// ------------------------------------------------------------------
// GEMM + bias: Y[r, c] = sum_k X[r, k] * W[k, c] + b[c]   (K = N = GD = 128)
// One wave per 16-row strip; 8 accumulator tiles (16x16) cover all 128 cols.
// Uses V_WMMA_F32_16X16X4_F32 (exact fp32, CDNA5 matrix pipe).
// ------------------------------------------------------------------
__global__ __launch_bounds__(256) void gat2_gemm_bias(
    const float* __restrict__ X, const float* __restrict__ W,
    const float* __restrict__ bias, float* __restrict__ Y, int nrows) {
  const int lane  = threadIdx.x & 31;
  const int wave  = threadIdx.x >> 5;
  const int strip = blockIdx.x * 8 + wave;       // wave-uniform
  if (strip * 16 >= nrows) return;               // keeps EXEC all-ones per wave
  const int m0 = strip * 16;
  const int lm = lane & 15;                      // M (A) / N (B,C) within tile
  const int lk = lane >> 4;                      // selects K pair {0,1} vs {2,3}

#if defined(__gfx1250__) && __has_builtin(__builtin_amdgcn_wmma_f32_16x16x4_f32)
  v8f acc[8] = {};
  const float* xrow = X + (size_t)(m0 + lm) * GD;
  for (int k = 0; k < GD; k += 4) {
    // A fragment 16x4: lane lm holds X[m0+lm, k+2*lk .. k+2*lk+1]
    v2f a = *(const v2f*)(xrow + k + 2 * lk);
#pragma unroll
    for (int t = 0; t < 8; ++t) {
      const int n0 = t * 16;
      // B fragment 4x16: lane lm = column n0+lm, VGPR j = K row k+2*lk+j
      v2f bf;
      bf.x = W[(size_t)(k + 2 * lk) * GD + n0 + lm];
      bf.y = W[(size_t)(k + 2 * lk + 1) * GD + n0 + lm];
      acc[t] = __builtin_amdgcn_wmma_f32_16x16x4_f32(
          false, a, false, bf, (short)0, acc[t], false, false);
    }
  }
#pragma unroll
  for (int t = 0; t < 8; ++t) {
    const int n0 = t * 16;
    const float bb = bias[n0 + lm];
#pragma unroll
    for (int r = 0; r < 8; ++r) {
      const int row = m0 + r + 8 * lk;          // C layout: VGPR r -> M=r / r+8
      Y[(size_t)row * GD + n0 + lm] = acc[t][r] + bb;
    }
  }
#else
  // scalar fallback (host pass / builtin missing) — same output mapping
  for (int t = 0; t < 8; ++t) {
    const int col = t * 16 + lm;
    for (int r = 0; r < 8; ++r) {
      const int row = m0 + r + 8 * lk;
      float s = bias[col];
      for (int k = 0; k < GD; ++k) s += X[(size_t)row * GD + k] * W[(size_t)k * GD + col];
      Y[(size_t)row * GD + col] = s;
    }
  }
#endif
}

// ------------------------------------------------------------------
// Edge phase kernels. One wave per edge; lane covers 4 consecutive channels
// (32 lanes x 4 = 128 = D). Self-loop edges are e >= E with src = dst = e - E.
// ------------------------------------------------------------------
__device__ __forceinline__ void gat2_edge_ids(const int* __restrict__ ei, int e,
                                              int E, int& src, int& dst) {
  if (e < E) { src = ei[e]; dst = ei[E + e]; }
  else       { src = e - E; dst = src; }
}

// ---- layer 1: 4 heads x 32 channels; head = lane>>3 ----
__global__ __launch_bounds__(256) void gat2_score_h4(
    const int* __restrict__ ei, const float* __restrict__ xl,
    const float* __restrict__ xr, const float* __restrict__ att,
    float* __restrict__ score, unsigned* __restrict__ smax, int E, int E2) {
  const int e = blockIdx.x * 8 + (threadIdx.x >> 5);
  if (e >= E2) return;
  const int lane = threadIdx.x & 31;
  int src, dst; gat2_edge_ids(ei, e, E, src, dst);
  const float4 l = *(const float4*)(xl + (size_t)src * GD + lane * 4);
  const float4 r = *(const float4*)(xr + (size_t)dst * GD + lane * 4);
  const float4 a = *(const float4*)(att + lane * 4);   // att1 flat == channel idx
  float s = 0.f, v;
  v = l.x + r.x; v = v > 0.f ? v : GAT_NEG_SLOPE * v; s += v * a.x;
  v = l.y + r.y; v = v > 0.f ? v : GAT_NEG_SLOPE * v; s += v * a.y;
  v = l.z + r.z; v = v > 0.f ? v : GAT_NEG_SLOPE * v; s += v * a.z;
  v = l.w + r.w; v = v > 0.f ? v : GAT_NEG_SLOPE * v; s += v * a.w;
  s += __shfl_xor(s, 1, 32);                           // reduce 8-lane head group
  s += __shfl_xor(s, 2, 32);
  s += __shfl_xor(s, 4, 32);
  if ((lane & 7) == 0) {
    const int h = lane >> 3;
    score[(size_t)e * 4 + h] = s;
    atomicMax(&smax[(size_t)dst * 4 + h], gat2_f2ord(s));
  }
}

__global__ void gat2_exp_h4(const int* __restrict__ ei, float* __restrict__ score,
                            const unsigned* __restrict__ smax,
                            float* __restrict__ denom, int E, int E2) {
  const int t = blockIdx.x * blockDim.x + threadIdx.x;
  if (t >= E2 * 4) return;
  const int e = t >> 2, h = t & 3;
  const int dst = (e < E) ? ei[E + e] : (e - E);
  const float m = gat2_ord2f(smax[(size_t)dst * 4 + h]);
  const float ex = expf(score[t] - m);
  score[t] = ex;                                       // overwrite with exp()
  atomicAdd(&denom[(size_t)dst * 4 + h], ex);
}

__global__ __launch_bounds__(256) void gat2_aggr_h4(
    const int* __restrict__ ei, const float* __restrict__ xl,
    const float* __restrict__ score, const float* __restrict__ denom,
    float* __restrict__ out, int E, int E2) {
  const int e = blockIdx.x * 8 + (threadIdx.x >> 5);
  if (e >= E2) return;
  const int lane = threadIdx.x & 31;
  int src, dst; gat2_edge_ids(ei, e, E, src, dst);
  const int h = lane >> 3;
  const float alpha = score[(size_t)e * 4 + h] /
                      (denom[(size_t)dst * 4 + h] + 1e-16f);
  const float4 l = *(const float4*)(xl + (size_t)src * GD + lane * 4);
  float* o = out + (size_t)dst * GD + lane * 4;
  atomicAdd(o + 0, l.x * alpha);
  atomicAdd(o + 1, l.y * alpha);
  atomicAdd(o + 2, l.z * alpha);
  atomicAdd(o + 3, l.w * alpha);
}

// ---- layer 2: single head over all 128 channels ----
__global__ __launch_bounds__(256) void gat2_score_h1(
    const int* __restrict__ ei, const float* __restrict__ xl,
    const float* __restrict__ xr, const float* __restrict__ att,
    float* __restrict__ score, unsigned* __restrict__ smax, int E, int E2) {
  const int e = blockIdx.x * 8 + (threadIdx.x >> 5);
  if (e >= E2) return;
  const int lane = threadIdx.x & 31;
  int src, dst; gat2_edge_ids(ei, e, E, src, dst);
  const float4 l = *(const float4*)(xl + (size_t)src * GD + lane * 4);
  const float4 r = *(const float4*)(xr + (size_t)dst * GD + lane * 4);
  const float4 a = *(const float4*)(att + lane * 4);
  float s = 0.f, v;
  v = l.x + r.x; v = v > 0.f ? v : GAT_NEG_SLOPE * v; s += v * a.x;
  v = l.y + r.y; v = v > 0.f ? v : GAT_NEG_SLOPE * v; s += v * a.y;
  v = l.z + r.z; v = v > 0.f ? v : GAT_NEG_SLOPE * v; s += v * a.z;
  v = l.w + r.w; v = v > 0.f ? v : GAT_NEG_SLOPE * v; s += v * a.w;
  s += __shfl_xor(s, 1, 32);                           // full-wave reduction
  s += __shfl_xor(s, 2, 32);
  s += __shfl_xor(s, 4, 32);
  s += __shfl_xor(s, 8, 32);
  s += __shfl_xor(s, 16, 32);
  if (lane == 0) {
    score[e] = s;
    atomicMax(&smax[dst], gat2_f2ord(s));
  }
}

__global__ void gat2_exp_h1(const int* __restrict__ ei, float* __restrict__ score,
                            const unsigned* __restrict__ smax,
                            float* __restrict__ denom, int E, int E2) {
  const int e = blockIdx.x * blockDim.x + threadIdx.x;
  if (e >= E2) return;
  const int dst = (e < E) ? ei[E + e] : (e - E);
  const float ex = expf(score[e] - gat2_ord2f(smax[dst]));
  score[e] = ex;
  atomicAdd(&denom[dst], ex);
}

__global__ __launch_bounds__(256) void gat2_aggr_h1(
    const int* __restrict__ ei, const float* __restrict__ xl,
    const float* __restrict__ score, const float* __restrict__ denom,
    float* __restrict__ out, int E, int E2) {
  const int e = blockIdx.x * 8 + (threadIdx.x >> 5);
  if (e >= E2) return;
  const int lane = threadIdx.x & 31;
  int src, dst; gat2_edge_ids(ei, e, E, src, dst);
  const float alpha = score[e] / (denom[dst] + 1e-16f);
  const float4 l = *(const float4*)(xl + (size_t)src * GD + lane * 4);
  float* o = out + (size_t)dst * GD + lane * 4;
  atomicAdd(o + 0, l.x * alpha);
  atomicAdd(o + 1, l.y * alpha);
  atomicAdd(o + 2, l.z * alpha);
  atomicAdd(o + 3, l.w * alpha);
}

// ---- elementwise epilogues ----
__global__ void gat2_bias_relu(float* __restrict__ x, const float* __restrict__ b, int n) {
  int t = blockIdx.x * blockDim.x + threadIdx.x;
  if (t < n) { float v = x[t] + b[t & (GD - 1)]; x[t] = v > 0.f ? v : 0.f; }
}
__global__ void gat2_bias_add(float* __restrict__ x, const float* __restrict__ b, int n) {
  int t = blockIdx.x * blockDim.x + threadIdx.x;
  if (t < n) x[t] += b[t & (GD - 1)];
}

// ------------------------------------------------------------------
extern "C" void kernel_launch(void* const* d_in, const int* in_sizes, int n_in,
                              void* d_out, int out_size, void* d_ws, size_t ws_size,
                              hipStream_t stream) {
  const int*   ei    = (const int*)d_in[0];
  const float* embed = (const float*)d_in[1];
  const float* Wl1   = (const float*)d_in[2];
  const float* bl1   = (const float*)d_in[3];
  const float* Wr1   = (const float*)d_in[4];
  const float* br1   = (const float*)d_in[5];
  const float* att1  = (const float*)d_in[6];
  const float* b1    = (const float*)d_in[7];
  const float* Wl2   = (const float*)d_in[8];
  const float* bl2   = (const float*)d_in[9];
  const float* Wr2   = (const float*)d_in[10];
  const float* br2   = (const float*)d_in[11];
  const float* att2  = (const float*)d_in[12];
  const float* b2    = (const float*)d_in[13];

  const int E  = in_sizes[0] / 2;
  const int N  = in_sizes[1] / GD;
  const int E2 = E + N;                          // edges + self loops

  // workspace layout (floats): xl | xr | xacc | score(E2*4) | smax(N*4 u32) | denom(N*4)
  float*    xl    = (float*)d_ws;
  float*    xr    = xl   + (size_t)N * GD;
  float*    xacc  = xr   + (size_t)N * GD;
  float*    score = xacc + (size_t)N * GD;
  unsigned* smax  = (unsigned*)(score + (size_t)E2 * 4);
  float*    denom = (float*)(smax + (size_t)N * 4);
  float*    out   = (float*)d_out;

  const dim3 blk(256);
  const int strips = (N + 15) / 16;
  const dim3 gemmGrid((strips + 7) / 8);
  const dim3 edgeGrid((E2 + 7) / 8);             // 8 waves / block, wave per edge
  const int nND = N * GD;

  // ---------------- layer 1 (4 heads x 32 ch) ----------------
  gat2_gemm_bias<<<gemmGrid, blk, 0, stream>>>(embed, Wl1, bl1, xl, N);
  gat2_gemm_bias<<<gemmGrid, blk, 0, stream>>>(embed, Wr1, br1, xr, N);
  gat2_fill_u32<<<(N * 4 + 255) / 256, blk, 0, stream>>>(smax, 0u, N * 4);
  gat2_zero_f32<<<(N * 4 + 255) / 256, blk, 0, stream>>>(denom, N * 4);
  gat2_zero_f32<<<(nND + 255) / 256, blk, 0, stream>>>(xacc, nND);
  gat2_score_h4<<<edgeGrid, blk, 0, stream>>>(ei, xl, xr, att1, score, smax, E, E2);
  gat2_exp_h4<<<(E2 * 4 + 255) / 256, blk, 0, stream>>>(ei, score, smax, denom, E, E2);
  gat2_aggr_h4<<<edgeGrid, blk, 0, stream>>>(ei, xl, score, denom, xacc, E, E2);
  gat2_bias_relu<<<(nND + 255) / 256, blk, 0, stream>>>(xacc, b1, nND);

  // ---------------- layer 2 (1 head x 128 ch) ----------------
  gat2_gemm_bias<<<gemmGrid, blk, 0, stream>>>(xacc, Wl2, bl2, xl, N);
  gat2_gemm_bias<<<gemmGrid, blk, 0, stream>>>(xacc, Wr2, br2, xr, N);
  gat2_fill_u32<<<(N + 255) / 256, blk, 0, stream>>>(smax, 0u, N);
  gat2_zero_f32<<<(N + 255) / 256, blk, 0, stream>>>(denom, N);
  gat2_zero_f32<<<(nND + 255) / 256, blk, 0, stream>>>(out, nND);
  gat2_score_h1<<<edgeGrid, blk, 0, stream>>>(ei, xl, xr, att2, score, smax, E, E2);
  gat2_exp_h1<<<(E2 + 255) / 256, blk, 0, stream>>>(ei, score, smax, denom, E, E2);
  gat2_aggr_h1<<<edgeGrid, blk, 0, stream>>>(ei, xl, score, denom, out, E, E2);
  gat2_bias_add<<<(nND + 255) / 256, blk, 0, stream>>>(out, b2, nND);
}